// Resize_17128329576938
// MI455X (gfx1250) — compile-verified
//
#include <hip/hip_runtime.h>

#ifndef __has_builtin
#define __has_builtin(x) 0
#endif

namespace {
constexpr int DIN  = 128;   // input spatial extent (all 3 axes)
constexpr int DOUT = 192;   // output spatial extent = DIN * 1.5
constexpr int NB   = 2;     // batch
}

__device__ __forceinline__ int imin_(int a, int b) { return a < b ? a : b; }

// One workgroup: output tile [3 z][3 y][192 x][2 c] from input tile [3][3][128][2] in LDS.
__global__ __launch_bounds__(192)
void resize_trilinear_1p5(const float* __restrict__ in, float* __restrict__ out) {
  const int t  = threadIdx.x;   // output x coordinate, 0..191
  const int by = blockIdx.x;    // output y tile index (3 rows each)
  const int bz = blockIdx.y;    // output z tile index (3 planes each)
  const int b  = blockIdx.z;    // batch

  // LDS tile: input [z:3][y:3][x:128] of float2 (the C=2 channel pair) = 9216 B
  __shared__ float2 tile[3 * 3 * DIN];

  const float2* __restrict__ in2 = reinterpret_cast<const float2*>(in);

  // ---- stage the 3x3x128 float2 input tile into LDS via async copies ----
  const int zin0 = 2 * bz;
  const int yin0 = 2 * by;
#pragma unroll
  for (int k = 0; k < 6; ++k) {
    const int e   = t + 192 * k;     // 0..1151 element of the tile
    const int z   = e / 384;
    const int rem = e - z * 384;
    const int y   = rem >> 7;        // /128
    const int x   = rem & 127;
    const int zi  = imin_(zin0 + z, DIN - 1);   // clamped halo
    const int yi  = imin_(yin0 + y, DIN - 1);
    const float2* src = &in2[(((size_t)b * DIN + zi) * DIN + yi) * DIN + x];
    unsigned int lds_addr =
        (unsigned int)(unsigned long long)(const void*)&tile[e];   // low 32b of flat LDS ptr
    unsigned long long gaddr = (unsigned long long)(const void*)src;
    asm volatile("global_load_async_to_lds_b64 %0, %1, off"
                 :: "v"(lds_addr), "v"(gaddr)
                 : "memory");
  }
#if __has_builtin(__builtin_amdgcn_s_wait_asynccnt)
  __builtin_amdgcn_s_wait_asynccnt(0);
#else
  asm volatile("s_wait_asynccnt 0" ::: "memory");
#endif
  __syncthreads();

  const float maxc = (float)(DIN - 1);

  // x-axis weights for this thread (exact neurite formula: loc = i/1.5, clipped)
  const float locx = (float)t / 1.5f;
  int ix0 = (int)floorf(locx); if (ix0 > DIN - 1) ix0 = DIN - 1;
  const int   ix1 = imin_(ix0 + 1, DIN - 1);
  const float wx0 = (float)ix1 - fminf(locx, maxc);
  const float wx1 = 1.0f - wx0;

  // x-interpolate the 9 (z,y) rows once -> registers
  float2 xv[9];
#pragma unroll
  for (int zy = 0; zy < 9; ++zy) {
    const float2 a = tile[zy * DIN + ix0];
    const float2 c = tile[zy * DIN + ix1];
    xv[zy].x = a.x * wx0 + c.x * wx1;
    xv[zy].y = a.y * wx0 + c.y * wx1;
  }

  // z / y lower-corner weights for the 3 outputs along each axis.
  // Local corner indices inside the tile are compile-time constants:
  //   j=0 -> (0,1), j=1 -> (0,1), j=2 -> (1,2)
  // (upper edge: clamped halo + w0==0 makes this exact too)
  float wz0[3], wy0[3];
#pragma unroll
  for (int j = 0; j < 3; ++j) {
    {
      const int o = 3 * bz + j;
      const float loc = (float)o / 1.5f;
      int i0 = (int)floorf(loc); if (i0 > DIN - 1) i0 = DIN - 1;
      const int i1 = imin_(i0 + 1, DIN - 1);
      wz0[j] = (float)i1 - fminf(loc, maxc);
    }
    {
      const int o = 3 * by + j;
      const float loc = (float)o / 1.5f;
      int i0 = (int)floorf(loc); if (i0 > DIN - 1) i0 = DIN - 1;
      const int i1 = imin_(i0 + 1, DIN - 1);
      wy0[j] = (float)i1 - fminf(loc, maxc);
    }
  }

  constexpr int L0[3] = {0, 0, 1};
  constexpr int L1[3] = {1, 1, 2};

  float2* __restrict__ out2 = reinterpret_cast<float2*>(out);
#pragma unroll
  for (int zz = 0; zz < 3; ++zz) {
#pragma unroll
    for (int yy = 0; yy < 3; ++yy) {
      const float2 v00 = xv[L0[zz] * 3 + L0[yy]];
      const float2 v01 = xv[L0[zz] * 3 + L1[yy]];
      const float2 v10 = xv[L1[zz] * 3 + L0[yy]];
      const float2 v11 = xv[L1[zz] * 3 + L1[yy]];
      const float a0 = wy0[yy], a1 = 1.0f - a0;
      const float c0 = wz0[zz], c1 = 1.0f - c0;
      float2 r;
      r.x = c0 * (a0 * v00.x + a1 * v01.x) + c1 * (a0 * v10.x + a1 * v11.x);
      r.y = c0 * (a0 * v00.y + a1 * v01.y) + c1 * (a0 * v10.y + a1 * v11.y);
      const size_t oidx =
          (((size_t)b * DOUT + (3 * bz + zz)) * DOUT + (3 * by + yy)) * DOUT + t;
      out2[oidx] = r;   // coalesced 192*8B burst per (zz,yy)
    }
  }
}

extern "C" void kernel_launch(void* const* d_in, const int* in_sizes, int n_in,
                              void* d_out, int out_size, void* d_ws, size_t ws_size,
                              hipStream_t stream) {
  (void)in_sizes; (void)n_in; (void)d_ws; (void)ws_size; (void)out_size;
  const float* in = (const float*)d_in[0];
  float* out = (float*)d_out;
  // grid: (y tiles, z tiles, batch); block: 192 threads = 6 wave32 waves
  dim3 grid(DOUT / 3, DOUT / 3, NB);
  resize_trilinear_1p5<<<grid, 192, 0, stream>>>(in, out);
}